// DirectMaxPlusAlphaMinPool2d_51316269253232
// MI455X (gfx1250) — compile-verified
//
#include <hip/hip_runtime.h>
#include <float.h>
#include <stdint.h>

// Problem geometry (fixed by the reference):
//   x: (32, 1600, 28, 28) fp32 -> 6400 rows of 6272 floats, out: 6400 fp32.
#define ROW_LEN      6272      // 28*28*8
#define CHUNK_FLOATS 128       // one async b128 per lane covers a chunk
#define CHUNK_BYTES  512
#define CHUNKS       49        // 6272 / 128
#define NBUF         4         // async pipeline depth (power of two)
#define WAVES        8         // waves (rows) per block
#define KSEL         20
#define ALPHA        0.7f

// ---- CDNA5 async global->LDS path (gfx1250) --------------------------------
// GVS mode: mem = SGPR64 base + VGPR32 offset; VDST VGPR holds the per-lane
// LDS byte address (low 32 bits of the flat shared pointer == group-relative
// LDS address per the gfx1250 aperture rules).
__device__ __forceinline__ void async_chunk_load(unsigned lds_addr, unsigned voff,
                                                 const float* base) {
  asm volatile("global_load_async_to_lds_b128 %0, %1, %2"
               :: "v"(lds_addr), "v"(voff),
                  "s"((unsigned long long)(uintptr_t)base)
               : "memory");
}

#define WAIT_ASYNC(n) asm volatile("s_wait_asynccnt " #n ::: "memory")
#define WAIT_DS0()    asm volatile("s_wait_dscnt 0" ::: "memory")

// ---- wave32 bitonic helpers ------------------------------------------------
// Full bitonic sort of 32 values (one per lane) into ASCENDING lane order.
__device__ __forceinline__ float sort32_asc(float v, int lane) {
  #pragma unroll
  for (int k = 2; k <= 32; k <<= 1) {
    #pragma unroll
    for (int j = k >> 1; j > 0; j >>= 1) {
      float p = __shfl_xor(v, j, 32);
      bool blockAsc = ((lane & k) == 0);
      bool upper    = ((lane & j) != 0);
      v = (upper == blockAsc) ? fmaxf(v, p) : fminf(v, p);
    }
  }
  return v;
}

// Bitonic cleanup: bitonic input -> DESCENDING lane order (lane 0 = largest).
__device__ __forceinline__ float bmerge32_desc(float v, int lane) {
  #pragma unroll
  for (int j = 16; j > 0; j >>= 1) {
    float p = __shfl_xor(v, j, 32);
    v = ((lane & j) != 0) ? fminf(v, p) : fmaxf(v, p);
  }
  return v;
}

__global__ __launch_bounds__(32 * WAVES)
void topk_pool_kernel(const float* __restrict__ x, float* __restrict__ out) {
  const int lane = threadIdx.x & 31;
  const int wv   = threadIdx.x >> 5;
  const int row  = blockIdx.x * WAVES + wv;
  const float* rowp = x + (size_t)row * ROW_LEN;

  extern __shared__ __align__(16) char smem[];
  char* wsm = smem + wv * (NBUF * CHUNK_BYTES);
  const unsigned lds_lane = (unsigned)(uintptr_t)wsm + (unsigned)(lane * 16);

  // Prologue: fill the async pipeline (NBUF chunks in flight).
  #pragma unroll
  for (int c = 0; c < NBUF; ++c)
    async_chunk_load(lds_lane + (unsigned)(c * CHUNK_BYTES),
                     (unsigned)(c * CHUNK_BYTES + lane * 16), rowp);

  // Running top-32 (Lt) and bottom-32 (Lb) of the row, one element per lane,
  // both kept sorted DESCENDING across lane index. Exact since KSEL <= 32.
  float Lt = -FLT_MAX, Lb = FLT_MAX;
  float tmin = -FLT_MAX;  // Lt[lane 31]: current 32nd-largest
  float bmax =  FLT_MAX;  // Lb[lane 0]:  current 32nd-smallest

  for (int c = 0; c < CHUNKS; ++c) {
    const int rem = (CHUNKS - 1) - c;  // chunks still needed after this one
    if (rem >= 3)      { WAIT_ASYNC(3); }
    else if (rem == 2) { WAIT_ASYNC(2); }
    else if (rem == 1) { WAIT_ASYNC(1); }
    else               { WAIT_ASYNC(0); }

    const float4 q =
        *(const float4*)(wsm + (c & (NBUF - 1)) * CHUNK_BYTES + lane * 16);

    const float vals[4] = {q.x, q.y, q.z, q.w};
    #pragma unroll
    for (int j = 0; j < 4; ++j) {
      const float v = vals[j];
      // Wave-uniform gate: scalar branch, no divergence.
      const bool qt = __ballot(v > tmin) != 0ull;
      const bool qb = __ballot(v < bmax) != 0ull;
      if (qt || qb) {
        const float s = sort32_asc(v, lane);  // batch ascending across lanes
        if (qt) {
          // top-32 of (Lt ∪ batch): halver max of desc vs asc, then cleanup.
          Lt = bmerge32_desc(fmaxf(Lt, s), lane);
          tmin = __shfl(Lt, 31, 32);
        }
        if (qb) {
          // bottom-32 of (Lb ∪ batch): halver min, then cleanup.
          Lb = bmerge32_desc(fminf(Lb, s), lane);
          bmax = __shfl(Lb, 0, 32);
        }
      }
    }

    const int nc = c + NBUF;
    if (nc < CHUNKS) {
      WAIT_DS0();  // ds reads of the recycled buffer retired before overwrite
      async_chunk_load(lds_lane + (unsigned)((nc & (NBUF - 1)) * CHUNK_BYTES),
                       (unsigned)(nc * CHUNK_BYTES + lane * 16), rowp);
    }
  }

  // Lt desc: 20 largest are lanes 0..19.  Lb desc: 20 smallest are lanes 12..31.
  float ts = (lane < KSEL) ? Lt : 0.0f;
  float bs = (lane >= 32 - KSEL) ? Lb : 0.0f;
  #pragma unroll
  for (int m = 16; m >= 1; m >>= 1) {
    ts += __shfl_xor(ts, m, 32);
    bs += __shfl_xor(bs, m, 32);
  }
  if (lane == 0)
    out[row] = 0.5f * (ts * (1.0f / KSEL) + ALPHA * (bs * (1.0f / KSEL)));
}

extern "C" void kernel_launch(void* const* d_in, const int* in_sizes, int n_in,
                              void* d_out, int out_size, void* d_ws, size_t ws_size,
                              hipStream_t stream) {
  (void)in_sizes; (void)n_in; (void)d_ws; (void)ws_size;
  const float* x  = (const float*)d_in[0];
  float* out      = (float*)d_out;

  const int rows   = out_size;          // 6400
  const int blocks = rows / WAVES;      // 800
  const size_t shmem = (size_t)WAVES * NBUF * CHUNK_BYTES;  // 16 KB

  topk_pool_kernel<<<blocks, 32 * WAVES, shmem, stream>>>(x, out);
}